// DCVQQuantizerEMA_17892833755576
// MI455X (gfx1250) — compile-verified
//
#include <hip/hip_runtime.h>

typedef __attribute__((ext_vector_type(2))) float v2f;
typedef __attribute__((ext_vector_type(4))) float v4f;
typedef __attribute__((ext_vector_type(8))) float v8f;

#define N_SUB   16
#define M_CODES 512
#define DS      8
#define T_TOK   32768          // B*H*W = 32*32*32
#define HW      1024           // H*W
#define D_CH    128
#define BETA    0.25f
#define DECAY   0.99f
#define EPSV    1e-5f

// output offsets (floats), return order: z_q, loss, indices, new_cb, new_ecs, new_ema_w
#define OFF_ZQ    0
#define OFF_LOSS  4194304
#define OFF_IDX   4194305
#define OFF_CB    4718593
#define OFF_ECS   4784129
#define OFF_EMAW  4792321

// workspace layout (floats)
#define WS_COUNTS 0            // N_SUB*M_CODES = 8192
#define WS_DW     8192         // N_SUB*M_CODES*DS = 65536
#define WS_LOSS   73728        // 1
#define WS_TOTAL  73729

__global__ __launch_bounds__(256) void vq_ws_init(float* __restrict__ ws) {
    int i = blockIdx.x * blockDim.x + threadIdx.x;
    if (i < WS_TOTAL) ws[i] = 0.0f;
}

// Main kernel: distances via fp32 WMMA, argmin, gather z_q, scatter counts/dw, loss partials.
// grid: (T_TOK/128, N_SUB), block: 256 (8 waves); wave w handles tokens [b*128 + w*16, +16)
__global__ __launch_bounds__(256) void vq_main(const float* __restrict__ z,
                                               const float* __restrict__ cb,
                                               float* __restrict__ out,
                                               float* __restrict__ ws) {
    __shared__ float cb_lds[M_CODES * DS];   // 16 KB: this subspace's codebook
    __shared__ float cbsq_lds[M_CODES];      // 2 KB : per-code squared norms
    __shared__ int   idx_lds[128];           // argmin per token in block

    const int tid = threadIdx.x;
    const int n   = blockIdx.y;
    const float* cbn = cb + n * (M_CODES * DS);

    // stage codebook (4096 floats) into LDS as float4
    {
        const v4f* src = (const v4f*)cbn;
        v4f*       dst = (v4f*)cb_lds;
        #pragma unroll
        for (int i = 0; i < (M_CODES * DS / 4) / 256; ++i)
            dst[tid + i * 256] = src[tid + i * 256];
    }
    __syncthreads();
    for (int m = tid; m < M_CODES; m += 256) {
        float s = 0.0f;
        #pragma unroll
        for (int k = 0; k < DS; ++k) { float v = cb_lds[m * DS + k]; s += v * v; }
        cbsq_lds[m] = s;
    }
    __syncthreads();

    const int wave = tid >> 5;
    const int lane = tid & 31;
    const int col  = lane & 15;      // token-row for A loads / code-col for B & C
    const int half = lane >> 4;      // K-half selector per ISA 16x4 fp32 layout

    const int t0 = blockIdx.x * 128 + wave * 16;
    const int tA = t0 + col;                                   // token whose row this lane loads
    const int zbA = (tA >> 10) * (D_CH * HW) + n * (DS * HW) + (tA & (HW - 1));

    // A operands: 16x4 fp32 tiles. lanes 0-15: K=0,1 (kk) ; lanes 16-31: K=2,3
    v2f a0, a1;
    a0.x = z[zbA + (2 * half + 0) * HW];
    a0.y = z[zbA + (2 * half + 1) * HW];
    a1.x = z[zbA + (2 * half + 4) * HW];
    a1.y = z[zbA + (2 * half + 5) * HW];

    float bestv[8];
    int   besti[8];
    #pragma unroll
    for (int j = 0; j < 8; ++j) { bestv[j] = 3.4e38f; besti[j] = 0; }

    // sweep all 512 codes in 16-wide tiles; dist ~ cb_sq[m] - 2*I[t,m] (z_sq drops out of argmin)
    #pragma unroll 4
    for (int mt = 0; mt < M_CODES / 16; ++mt) {
        const int c = mt * 16 + col;
        const float* br = cb_lds + c * DS;
        v2f b0, b1;
        b0.x = br[2 * half + 0];
        b0.y = br[2 * half + 1];
        b1.x = br[2 * half + 4];
        b1.y = br[2 * half + 5];

        v8f acc = {};
        acc = __builtin_amdgcn_wmma_f32_16x16x4_f32(false, a0, false, b0,
                                                    (short)0, acc, false, false);
        acc = __builtin_amdgcn_wmma_f32_16x16x4_f32(false, a1, false, b1,
                                                    (short)0, acc, false, false);

        const float sc = cbsq_lds[c];
        #pragma unroll
        for (int j = 0; j < 8; ++j) {
            float d = sc - 2.0f * acc[j];
            if (d < bestv[j]) { bestv[j] = d; besti[j] = c; }   // strictly < keeps lowest index per lane
        }
    }

    // cross-lane argmin within each 16-lane half (rows 0-7 in lanes 0-15, rows 8-15 in lanes 16-31)
    #pragma unroll
    for (int j = 0; j < 8; ++j) {
        float v = bestv[j];
        int   i = besti[j];
        #pragma unroll
        for (int off = 8; off >= 1; off >>= 1) {
            float ov = __shfl_xor(v, off, 32);
            int   oi = __shfl_xor(i, off, 32);
            if (ov < v || (ov == v && oi < i)) { v = ov; i = oi; }
        }
        besti[j] = i;
    }
    if (col == 0) {
        #pragma unroll
        for (int j = 0; j < 8; ++j) idx_lds[wave * 16 + half * 8 + j] = besti[j];
    }
    __syncthreads();

    // post phase: lane -> (token = col, k-range = half*4..half*4+3)
    const int myidx = idx_lds[wave * 16 + col];
    const int tt = t0 + col;
    const int zb = (tt >> 10) * (D_CH * HW) + n * (DS * HW) + (tt & (HW - 1));
    const float* crow = cb_lds + myidx * DS;

    float lpart = 0.0f;
    #pragma unroll
    for (int kk = 0; kk < 4; ++kk) {
        const int k = half * 4 + kk;
        const float q  = crow[k];
        const float zz = z[zb + k * HW];
        out[OFF_ZQ + zb + k * HW] = q;                 // straight-through fwd == z_q, same layout as z
        const float df = q - zz;
        lpart += df * df;
        atomicAdd(&ws[WS_DW + (n * M_CODES + myidx) * DS + k], zz);
    }
    if (half == 0) {
        out[OFF_IDX + tt * N_SUB + n] = (float)myidx;  // indices reshaped [B,H,W,N] = t*N + n
        atomicAdd(&ws[WS_COUNTS + n * M_CODES + myidx], 1.0f);
    }
    // wave-reduce loss partial, one atomic per wave
    #pragma unroll
    for (int off = 16; off >= 1; off >>= 1) lpart += __shfl_xor(lpart, off, 32);
    if (lane == 0) atomicAdd(&ws[WS_LOSS], lpart);
}

// EMA finalize: one block per subspace, thread m handles code m
__global__ __launch_bounds__(512) void vq_ema(const float* __restrict__ ecs,
                                              const float* __restrict__ emaw,
                                              const float* __restrict__ ws,
                                              float* __restrict__ out) {
    __shared__ float red[M_CODES];
    const int n = blockIdx.x;
    const int m = threadIdx.x;

    const float cnt = ws[WS_COUNTS + n * M_CODES + m];
    const float ne  = ecs[n * M_CODES + m] * DECAY + (1.0f - DECAY) * cnt;
    red[m] = ne;
    __syncthreads();
    #pragma unroll
    for (int s = M_CODES / 2; s > 0; s >>= 1) {
        if (m < s) red[m] += red[m + s];
        __syncthreads();
    }
    const float nsum = red[0];
    const float cs = (ne + EPSV) / (nsum + (float)M_CODES * EPSV) * nsum;

    out[OFF_ECS + n * M_CODES + m] = ne;
    #pragma unroll
    for (int k = 0; k < DS; ++k) {
        const int i = (n * M_CODES + m) * DS + k;
        const float w = emaw[i] * DECAY + (1.0f - DECAY) * ws[WS_DW + i];
        out[OFF_EMAW + i] = w;
        out[OFF_CB + i]   = w / cs;
    }
}

__global__ void vq_loss(const float* __restrict__ ws, float* __restrict__ out) {
    out[OFF_LOSS] = BETA * ws[WS_LOSS] / (float)(T_TOK * N_SUB * DS);
}

extern "C" void kernel_launch(void* const* d_in, const int* in_sizes, int n_in,
                              void* d_out, int out_size, void* d_ws, size_t ws_size,
                              hipStream_t stream) {
    (void)in_sizes; (void)n_in; (void)out_size; (void)ws_size;
    const float* z    = (const float*)d_in[0];
    const float* cbk  = (const float*)d_in[1];
    const float* ecs  = (const float*)d_in[2];
    const float* emaw = (const float*)d_in[3];
    float* out = (float*)d_out;
    float* ws  = (float*)d_ws;

    vq_ws_init<<<(WS_TOTAL + 255) / 256, 256, 0, stream>>>(ws);
    dim3 grid(T_TOK / 128, N_SUB);
    vq_main<<<grid, 256, 0, stream>>>(z, cbk, out, ws);
    vq_ema<<<N_SUB, M_CODES, 0, stream>>>(ecs, emaw, ws, out);
    vq_loss<<<1, 1, 0, stream>>>(ws, out);
}